// RetinaFace_74474732912668
// MI455X (gfx1250) — compile-verified
//
#include <hip/hip_runtime.h>
#include <hip/hip_bf16.h>

// RetinaFace post-processing for MI455X (gfx1250, wave32).
// decode+softmax -> O(N^2) rank sort (LDS tiles staged with CDNA5 async
// global->LDS b128 loads, ASYNCcnt) -> batched greedy NMS (single WG,
// LDS-staged batches, leader wave resolves the serial dependency) -> writeout.

#define NN        16800
#define V0        0.1f
#define V1        0.2f
#define IOU_TH    0.4f
#define SCORE_TH  0.02f
#define NMS_B     128
#define NMS_T     1024
#define BLK       256
#define RTILE     1024     // score tile (floats) staged in LDS per pass

// ---------------------------------------------------------------- kernel 1
// softmax + decode: dec[i][16] = [x1,y1,x2,y2, lm0..lm9, 1.0, conf]
__global__ void k_decode(const float* __restrict__ bbox,
                         const float* __restrict__ landm,
                         const float* __restrict__ cls,
                         const float* __restrict__ pri,
                         float* __restrict__ dec,
                         float* __restrict__ scores, int n)
{
    int i = blockIdx.x * blockDim.x + threadIdx.x;
    if (i >= n) return;

    // 2-class softmax with max subtraction (matches jax.nn.softmax)
    float c0 = cls[i * 2 + 0], c1 = cls[i * 2 + 1];
    float m  = fmaxf(c0, c1);
    float e0 = expf(c0 - m), e1 = expf(c1 - m);
    float conf = e1 / (e0 + e1);

    float px = pri[i * 4 + 0], py = pri[i * 4 + 1];
    float pw = pri[i * 4 + 2], ph = pri[i * 4 + 3];

    float bx = bbox[i * 4 + 0], by = bbox[i * 4 + 1];
    float bw = bbox[i * 4 + 2], bh = bbox[i * 4 + 3];

    float cx = px + bx * V0 * pw;
    float cy = py + by * V0 * ph;
    float sw = pw * expf(bw * V1);
    float sh = ph * expf(bh * V1);

    float* row = dec + (size_t)i * 16;
    row[0] = cx - sw * 0.5f;
    row[1] = cy - sh * 0.5f;
    row[2] = cx + sw * 0.5f;
    row[3] = cy + sh * 0.5f;

    const float* lm = landm + (size_t)i * 10;
#pragma unroll
    for (int k = 0; k < 5; ++k) {
        row[4 + 2 * k]     = px + lm[2 * k]     * V0 * pw;
        row[4 + 2 * k + 1] = py + lm[2 * k + 1] * V0 * ph;
    }
    row[14] = 1.0f;
    row[15] = conf;
    scores[i] = conf;
}

// ---------------------------------------------------------------- kernel 2
// rank_i = #{j: s_j > s_i} + #{j<i: s_j == s_i}  (stable argsort(-s) position)
// then scatter decoded row i to sorted position rank_i.
// 1024-float score tiles staged into LDS with async global->LDS b128 loads.
__global__ void __launch_bounds__(BLK) k_rank_scatter(
        const float* __restrict__ scores,
        const float* __restrict__ dec,
        float* __restrict__ dec_s, int n)
{
    __shared__ __align__(16) float tile[RTILE];
    int tid = threadIdx.x;
    int i   = blockIdx.x * BLK + tid;
    float si = (i < n) ? scores[i] : 0.0f;
    int rank = 0;

    for (int base = 0; base < n; base += RTILE) {
        __syncthreads();                    // protect previous tile
        int j0 = base + tid * 4;
        if (j0 + 3 < n) {
            // CDNA5 async global->LDS (GVS: lds_vdst, 32b voffset, sgpr base)
            unsigned ldsoff = (unsigned)(unsigned long long)(void*)&tile[tid * 4];
            unsigned goff   = (unsigned)(j0 * 4);
            asm volatile("global_load_async_to_lds_b128 %0, %1, %2"
                         :: "v"(ldsoff), "v"(goff), "s"(scores) : "memory");
        } else {
#pragma unroll
            for (int e = 0; e < 4; ++e)     // ragged tail: plain LDS stores
                if (j0 + e < n) tile[tid * 4 + e] = scores[j0 + e];
        }
        if (base + 2 * RTILE < n)
            __builtin_prefetch(scores + base + 2 * RTILE, 0, 0);
        asm volatile("s_wait_asynccnt 0" ::: "memory");
        __syncthreads();

        int cnt = min(RTILE, n - base);
        if (i < n) {
            int c4 = cnt >> 2;
            const float4* t4 = (const float4*)tile;
            for (int t = 0; t < c4; ++t) {
                float4 v = t4[t];
                int jj = base + t * 4;
                rank += (v.x > si) || (v.x == si && (jj    ) < i);
                rank += (v.y > si) || (v.y == si && (jj + 1) < i);
                rank += (v.z > si) || (v.z == si && (jj + 2) < i);
                rank += (v.w > si) || (v.w == si && (jj + 3) < i);
            }
            for (int t = c4 * 4; t < cnt; ++t) {
                float sj = tile[t];
                int jj = base + t;
                rank += (sj > si) || (sj == si && jj < i);
            }
        }
    }

    if (i < n) {
        const float* src = dec   + (size_t)i    * 16;
        float*       dst = dec_s + (size_t)rank * 16;
#pragma unroll
        for (int c = 0; c < 16; ++c) dst[c] = src[c];
    }
}

// ---------------------------------------------------------------- kernel 3
__device__ __forceinline__ bool iou_gt(float x1, float y1, float x2, float y2,
                                       float a,
                                       float u1, float v1c, float u2, float v2,
                                       float b)
{
    float iw = fmaxf(fminf(x2, u2) - fmaxf(x1, u1), 0.0f);
    float ih = fmaxf(fminf(y2, v2) - fmaxf(y1, v1c), 0.0f);
    float inter = iw * ih;
    float iou = inter / (a + b - inter);
    return iou > IOU_TH;
}

// Batched greedy NMS, single workgroup. Batch (box/area/score) is staged in
// LDS by one thread per box (up to 128), then the leader wave (lockstep wave32) resolves
// the intra-batch serial dependency purely out of LDS (short latency chain);
// finally all threads suppress the O(N) tail in parallel.
__global__ void __launch_bounds__(NMS_T) k_nms(const float* __restrict__ dec_s,
                                               int* __restrict__ keep, int n)
{
    __shared__ unsigned char sup_[NN];
    __shared__ float bx1[NMS_B], by1[NMS_B], bx2[NMS_B], by2[NMS_B];
    __shared__ float bar[NMS_B], bsc[NMS_B];
    __shared__ float4 kbox[NMS_B];
    __shared__ float  karea[NMS_B];
    __shared__ int    kcnt;
    volatile unsigned char* sup = sup_;

    int tid = threadIdx.x;
    for (int t = tid; t < n; t += NMS_T) sup_[t] = 0;
    __syncthreads();

    for (int base = 0; base < n; base += NMS_B) {
        int bend = min(base + NMS_B, n);
        int bn   = bend - base;

        if (tid < bn) {                 // stage batch into LDS
            const float* r = dec_s + (size_t)(base + tid) * 16;
            float4 b = *(const float4*)r;          // rows are 64B-aligned
            bx1[tid] = b.x;  by1[tid] = b.y;
            bx2[tid] = b.z;  by2[tid] = b.w;
            bar[tid] = (b.z - b.x) * (b.w - b.y);
            bsc[tid] = r[15];
        }
        __syncthreads();

        if (tid < 32) {                 // leader wave: intra-batch greedy
            int cnt = 0;                // uniform across lanes
            for (int ii = 0; ii < bn; ++ii) {
                int i = base + ii;
                bool keep_i = (bsc[ii] >= SCORE_TH) && (sup[i] == 0);
                if (tid == 0) keep[i] = keep_i ? 1 : 0;
                if (keep_i) {
                    float x1 = bx1[ii], y1 = by1[ii];
                    float x2 = bx2[ii], y2 = by2[ii];
                    float a  = bar[ii];
                    if (tid == 0) {
                        kbox[cnt]  = make_float4(x1, y1, x2, y2);
                        karea[cnt] = a;
                    }
                    cnt++;
                    for (int jj = ii + 1 + tid; jj < bn; jj += 32) {
                        if (sup[base + jj]) continue;
                        if (iou_gt(x1, y1, x2, y2, a,
                                   bx1[jj], by1[jj], bx2[jj], by2[jj], bar[jj]))
                            sup[base + jj] = 1;
                    }
                }
                __builtin_amdgcn_wave_barrier();   // keep LDS ops ordered
            }
            if (tid == 0) kcnt = cnt;
        }
        __syncthreads();

        int cnt = kcnt;
        if (cnt > 0) {                  // all threads: suppress the tail
            for (int j = bend + tid; j < n; j += NMS_T) {
                if (sup[j]) continue;
                const float* rj = dec_s + (size_t)j * 16;
                float4 bj = *(const float4*)rj;
                float b = (bj.z - bj.x) * (bj.w - bj.y);
                for (int k = 0; k < cnt; ++k) {
                    float4 bb = kbox[k];
                    if (iou_gt(bb.x, bb.y, bb.z, bb.w, karea[k],
                               bj.x, bj.y, bj.z, bj.w, b)) { sup[j] = 1; break; }
                }
            }
        }
        __syncthreads();
    }
}

// ---------------------------------------------------------------- kernel 4
__global__ void k_writeout(const float* __restrict__ dec_s,
                           const int* __restrict__ keep,
                           float* __restrict__ out, int n)
{
    int i = blockIdx.x * blockDim.x + threadIdx.x;
    int total = n * 16;
    if (i < total) {
        int r = i >> 4;
        out[i] = keep[r] ? dec_s[i] : 0.0f;
    }
}

// ----------------------------------------------------------------- launch
extern "C" void kernel_launch(void* const* d_in, const int* in_sizes, int n_in,
                              void* d_out, int out_size, void* d_ws, size_t ws_size,
                              hipStream_t stream)
{
    const float* bbox  = (const float*)d_in[0];   // (1,N,4)
    const float* landm = (const float*)d_in[1];   // (1,N,10)
    const float* cls   = (const float*)d_in[2];   // (1,N,2)
    const float* pri   = (const float*)d_in[3];   // (N,4)

    int n = in_sizes[3] / 4;                      // N = 16800

    // workspace layout (floats): dec[N*16] | scores[N] | dec_s[N*16] | keep[N]
    float* ws     = (float*)d_ws;
    float* dec    = ws;
    float* scores = ws + (size_t)n * 16;
    float* dec_s  = ws + (size_t)n * 17;
    int*   keep   = (int*)(ws + (size_t)n * 33);

    int blocks = (n + BLK - 1) / BLK;
    k_decode<<<blocks, BLK, 0, stream>>>(bbox, landm, cls, pri, dec, scores, n);
    k_rank_scatter<<<blocks, BLK, 0, stream>>>(scores, dec, dec_s, n);
    k_nms<<<1, NMS_T, 0, stream>>>(dec_s, keep, n);
    k_writeout<<<(n * 16 + 255) / 256, 256, 0, stream>>>(dec_s, keep, (float*)d_out, n);
}